// BiquadResonator_80393197846992
// MI455X (gfx1250) — compile-verified
//
#include <hip/hip_runtime.h>
#include <stdint.h>

// Problem geometry (fixed by the reference)
#define BATCH   32
#define TLEN    262144
#define CTRL    1024
#define CHUNKS  256                  // samples per chunk == T/C control segment
#define NCHROW  1024                 // chunks per row (== CTRL)
#define NC      (BATCH * NCHROW)     // 32768 chunks total
#define NPAIR   (NC / 2)             // 16384 level-1 pair transforms

typedef float v2f __attribute__((ext_vector_type(2)));
typedef float v8f __attribute__((ext_vector_type(8)));

// ---------------------------------------------------------------------------
// CDNA5 async-to-LDS staging (ASYNCcnt path). 16 floats (4 x b128) per tile.
// ---------------------------------------------------------------------------
__device__ __forceinline__ void async_tile16(uint32_t lds_byte, const float* g) {
  asm volatile(
      "global_load_async_to_lds_b128 %0, %1, off\n\t"
      "global_load_async_to_lds_b128 %0, %1, off offset:16\n\t"
      "global_load_async_to_lds_b128 %0, %1, off offset:32\n\t"
      "global_load_async_to_lds_b128 %0, %1, off offset:48"
      :: "v"(lds_byte), "v"((unsigned long long)(uintptr_t)g)
      : "memory");
}
__device__ __forceinline__ void wait_async_le4() {
  asm volatile("s_wait_asynccnt 0x4" ::: "memory");   // current tile done, next in flight
}
__device__ __forceinline__ void wait_async_0() {
  asm volatile("s_wait_asynccnt 0x0" ::: "memory");
}

// ---------------------------------------------------------------------------
// Per-chunk control constants and per-sample biquad coefficients
// ---------------------------------------------------------------------------
struct Ctl { float fn0, dfn, bn0, dbn, g0, dg; };

__device__ __forceinline__ Ctl load_ctl(const float* __restrict__ freq,
                                        const float* __restrict__ bw,
                                        const float* __restrict__ gain,
                                        int row, int chunk) {
  int nxt = (chunk + 1 < CTRL) ? chunk + 1 : CTRL - 1;
  int i0 = row * CTRL + chunk, i1 = row * CTRL + nxt;
  const float inv_ny = 1.0f / 22050.0f;
  Ctl c;
  float f0 = fminf(fmaxf(freq[i0] * inv_ny, 1e-5f), 0.95f);
  float f1 = fminf(fmaxf(freq[i1] * inv_ny, 1e-5f), 0.95f);
  float b0 = fminf(fmaxf(bw[i0] * inv_ny, 1e-5f), 0.95f);
  float b1 = fminf(fmaxf(bw[i1] * inv_ny, 1e-5f), 0.95f);
  c.fn0 = f0; c.dfn = f1 - f0;
  c.bn0 = b0; c.dbn = b1 - b0;
  c.g0 = gain[i0]; c.dg = gain[i1] - gain[i0];
  return c;
}

__device__ __forceinline__ void biquad_coef(const Ctl& c, float frac,
                                            float& b0, float& a1, float& a2) {
  float f  = fmaf(c.dfn, frac, c.fn0);
  float bb = fmaf(c.dbn, frac, c.bn0);
  float g  = fmaf(c.dg,  frac, c.g0);
  float omega = 6.28318530717958647f * f;
  float s, co;
  __sincosf(omega, &s, &co);
  float z  = 0.346573590279972655f * bb * omega * __builtin_amdgcn_rcpf(s);
  float ez = __expf(z);
  float alpha = s * 0.5f * (ez - __builtin_amdgcn_rcpf(ez));   // sin*sinh(z)
  float ia0 = __builtin_amdgcn_rcpf(1.0f + alpha);
  b0 = alpha * g * ia0;
  a1 = -2.0f * co * ia0;
  a2 = (1.0f - alpha) * ia0;
}

// ---------------------------------------------------------------------------
// Affine 2x2 transform helpers: s_out = M * s_in + v ; compose(h,l) = h after l
// ---------------------------------------------------------------------------
struct Aff { float m00, m01, m10, m11, v0, v1; };

__device__ __forceinline__ Aff compose(const Aff& h, const Aff& l) {
  Aff r;
  r.m00 = h.m00 * l.m00 + h.m01 * l.m10;
  r.m01 = h.m00 * l.m01 + h.m01 * l.m11;
  r.m10 = h.m10 * l.m00 + h.m11 * l.m10;
  r.m11 = h.m10 * l.m01 + h.m11 * l.m11;
  r.v0  = h.m00 * l.v0  + h.m01 * l.v1 + h.v0;
  r.v1  = h.m10 * l.v0  + h.m11 * l.v1 + h.v1;
  return r;
}

// ---------------------------------------------------------------------------
// K1: per-chunk affine transform from zero state.  One thread = 256 samples.
// ---------------------------------------------------------------------------
__global__ __launch_bounds__(256) void k1_chunk_transforms(
    const float* __restrict__ x, const float* __restrict__ freq,
    const float* __restrict__ bw, const float* __restrict__ gain,
    float* __restrict__ w) {
  __shared__ float lds[256 * 36];          // 36 floats/thread: 2x16 dbl-buf + pad
  int tid = blockIdx.x * 256 + threadIdx.x;
  int row = tid >> 10, chunk = tid & (NCHROW - 1);
  const float* gx = x + (size_t)row * TLEN + (size_t)chunk * CHUNKS;
  uint32_t lds_byte = (uint32_t)(uintptr_t)&lds[threadIdx.x * 36];

  float x1 = 0.f, x2 = 0.f;
  if (chunk > 0) { x1 = gx[-1]; x2 = gx[-2]; }
  async_tile16(lds_byte, gx);              // prefetch tile 0

  Ctl ctl = load_ctl(freq, bw, gain, row, chunk);
  float M00 = 1.f, M01 = 0.f, M10 = 0.f, M11 = 1.f, v0 = 0.f, v1 = 0.f;

  for (int tile = 0; tile < 16; ++tile) {
    int buf = tile & 1;
    if (tile < 15) { async_tile16(lds_byte + (buf ^ 1) * 64u, gx + (tile + 1) * 16); wait_async_le4(); }
    else           { wait_async_0(); }
#pragma unroll
    for (int i = 0; i < 16; ++i) {
      float xt = lds[threadIdx.x * 36 + buf * 16 + i];
      float frac = (float)(tile * 16 + i) * (1.0f / 256.0f);
      float b0, a1, a2;
      biquad_coef(ctl, frac, b0, a1, a2);
      float u  = b0 * (xt - x2);           // b2 == -b0
      float nv = u - a1 * v0 - a2 * v1;    // v <- A*v + e*u
      v1 = v0; v0 = nv;
      float n0 = -(a1 * M00 + a2 * M10);   // M <- A*M, A = [[-a1,-a2],[1,0]]
      float n1 = -(a1 * M01 + a2 * M11);
      M10 = M00; M11 = M01; M00 = n0; M01 = n1;
      x2 = x1; x1 = xt;
    }
  }
  w[0 * NC + tid] = M00; w[1 * NC + tid] = M01;
  w[2 * NC + tid] = M10; w[3 * NC + tid] = M11;
  w[4 * NC + tid] = v0;  w[5 * NC + tid] = v1;
}

// ---------------------------------------------------------------------------
// K2a: pairwise composition 1024 -> 512 transforms/row.  Each wave composes 2
// pairs via one v_wmma_f32_16x16x4_f32: A = blockdiag(M_hi), B = [M_lo | v_lo].
// ---------------------------------------------------------------------------
__global__ __launch_bounds__(256) void k2a_pair_compose(float* __restrict__ w) {
  const float* M00 = w + 0 * NC; const float* M01 = w + 1 * NC;
  const float* M10 = w + 2 * NC; const float* M11 = w + 3 * NC;
  const float* V0  = w + 4 * NC; const float* V1  = w + 5 * NC;
  float* R00 = w + 6 * NC + 0 * NPAIR; float* R01 = w + 6 * NC + 1 * NPAIR;
  float* R10 = w + 6 * NC + 2 * NPAIR; float* R11 = w + 6 * NC + 3 * NPAIR;
  float* RV0 = w + 6 * NC + 4 * NPAIR; float* RV1 = w + 6 * NC + 5 * NPAIR;

  int gt = blockIdx.x * 256 + threadIdx.x;
  int wave = gt >> 5, lane = gt & 31;
  int gp0 = wave * 2, gp1 = gp0 + 1;
  int p0lo = ((gp0 >> 9) << 10) + ((gp0 & 511) << 1), p0hi = p0lo + 1;
  int p1lo = ((gp1 >> 9) << 10) + ((gp1 & 511) << 1), p1hi = p1lo + 1;

#if __has_builtin(__builtin_amdgcn_wmma_f32_16x16x4_f32)
  // f32 16x16x4 layouts (ISA 7.12.2): A[m][k]: lanes0-15 k={0,1} in v0/v1,
  // lanes16-31 k={2,3}.  B[k][n]: v0 = rows 0/2, v1 = rows 1/3 (lane halves).
  // D[v]: lanes0-15 row v, lanes16-31 row v+8.
  int l16 = lane & 15;
  bool hihalf = lane >= 16;
  float ax = 0.f, ay = 0.f, bx = 0.f, by = 0.f;
  if (!hihalf) {                                 // pair0 block: rows/cols 0-1, B cols 0-2
    if (l16 == 0)      { ax = M00[p0hi]; ay = M01[p0hi]; bx = M00[p0lo]; by = M10[p0lo]; }
    else if (l16 == 1) { ax = M10[p0hi]; ay = M11[p0hi]; bx = M01[p0lo]; by = M11[p0lo]; }
    else if (l16 == 2) { bx = V0[p0lo]; by = V1[p0lo]; }
  } else {                                       // pair1 block: rows/cols 2-3, B cols 3-5
    if (l16 == 2)      { ax = M00[p1hi]; ay = M01[p1hi]; }
    else if (l16 == 3) { ax = M10[p1hi]; ay = M11[p1hi]; bx = M00[p1lo]; by = M10[p1lo]; }
    else if (l16 == 4) { bx = M01[p1lo]; by = M11[p1lo]; }
    else if (l16 == 5) { bx = V0[p1lo]; by = V1[p1lo]; }
  }
  v2f Af = { ax, ay };
  v2f Bf = { bx, by };
  v8f Cf = {};
  v8f D = __builtin_amdgcn_wmma_f32_16x16x4_f32(false, Af, false, Bf, (short)0, Cf, false, false);
  if (!hihalf) {
    if (l16 == 0)      { R00[gp0] = D[0]; R10[gp0] = D[1]; }
    else if (l16 == 1) { R01[gp0] = D[0]; R11[gp0] = D[1]; }
    else if (l16 == 2) { RV0[gp0] = D[0] + V0[p0hi]; RV1[gp0] = D[1] + V1[p0hi]; }
    else if (l16 == 3) { R00[gp1] = D[2]; R10[gp1] = D[3]; }
    else if (l16 == 4) { R01[gp1] = D[2]; R11[gp1] = D[3]; }
    else if (l16 == 5) { RV0[gp1] = D[2] + V0[p1hi]; RV1[gp1] = D[3] + V1[p1hi]; }
  }
#else
  if (lane < 2) {
    int gp = gp0 + lane;
    int lo = lane ? p1lo : p0lo, hh = lane ? p1hi : p0hi;
    Aff h = { M00[hh], M01[hh], M10[hh], M11[hh], V0[hh], V1[hh] };
    Aff l = { M00[lo], M01[lo], M10[lo], M11[lo], V0[lo], V1[lo] };
    Aff r = compose(h, l);
    R00[gp] = r.m00; R01[gp] = r.m01; R10[gp] = r.m10; R11[gp] = r.m11;
    RV0[gp] = r.v0;  RV1[gp] = r.v1;
  }
#endif
}

// ---------------------------------------------------------------------------
// K2b: one wave per row.  Lane-serial combine of 16 level-1 transforms, wave32
// shuffle scan, then emit per-chunk entry states (y1,y2) by walking level-0.
// ---------------------------------------------------------------------------
__global__ __launch_bounds__(32) void k2b_scan(float* __restrict__ w) {
  const float* M00 = w + 0 * NC; const float* M01 = w + 1 * NC;
  const float* M10 = w + 2 * NC; const float* M11 = w + 3 * NC;
  const float* V0  = w + 4 * NC; const float* V1  = w + 5 * NC;
  const float* R00 = w + 6 * NC + 0 * NPAIR; const float* R01 = w + 6 * NC + 1 * NPAIR;
  const float* R10 = w + 6 * NC + 2 * NPAIR; const float* R11 = w + 6 * NC + 3 * NPAIR;
  const float* RV0 = w + 6 * NC + 4 * NPAIR; const float* RV1 = w + 6 * NC + 5 * NPAIR;
  float* SY1 = w + 6 * NC + 6 * NPAIR;   // = w + 9*NC
  float* SY2 = SY1 + NC;

  int row = blockIdx.x, lane = threadIdx.x;
  int pbase = row * (NCHROW / 2) + lane * 16;

  Aff P = { R00[pbase], R01[pbase], R10[pbase], R11[pbase], RV0[pbase], RV1[pbase] };
  for (int j = 1; j < 16; ++j) {
    Aff t = { R00[pbase + j], R01[pbase + j], R10[pbase + j], R11[pbase + j],
              RV0[pbase + j], RV1[pbase + j] };
    P = compose(t, P);
  }
  for (int d = 1; d < 32; d <<= 1) {        // wave32 inclusive scan
    Aff q;
    q.m00 = __shfl_up(P.m00, (unsigned)d, 32); q.m01 = __shfl_up(P.m01, (unsigned)d, 32);
    q.m10 = __shfl_up(P.m10, (unsigned)d, 32); q.m11 = __shfl_up(P.m11, (unsigned)d, 32);
    q.v0  = __shfl_up(P.v0,  (unsigned)d, 32); q.v1  = __shfl_up(P.v1,  (unsigned)d, 32);
    if (lane >= d) P = compose(P, q);
  }
  Aff E;                                     // exclusive prefix
  E.m00 = __shfl_up(P.m00, 1u, 32); E.m01 = __shfl_up(P.m01, 1u, 32);
  E.m10 = __shfl_up(P.m10, 1u, 32); E.m11 = __shfl_up(P.m11, 1u, 32);
  E.v0  = __shfl_up(P.v0,  1u, 32); E.v1  = __shfl_up(P.v1,  1u, 32);
  if (lane == 0) { E = Aff{1.f, 0.f, 0.f, 1.f, 0.f, 0.f}; }

  float s0 = E.v0, s1 = E.v1;                // s_init == 0 -> state = E.v
  int cbase = row * NCHROW + lane * 32;
  for (int k = 0; k < 32; ++k) {
    int c = cbase + k;
    SY1[c] = s0; SY2[c] = s1;                // entry state (y1,y2) for chunk c
    float n0 = M00[c] * s0 + M01[c] * s1 + V0[c];
    float n1 = M10[c] * s0 + M11[c] * s1 + V1[c];
    s0 = n0; s1 = n1;
  }
}

// ---------------------------------------------------------------------------
// K3: re-run each chunk with its correct entry state, write y and carries.
// ---------------------------------------------------------------------------
__global__ __launch_bounds__(256) void k3_finalize(
    const float* __restrict__ x, const float* __restrict__ freq,
    const float* __restrict__ bw, const float* __restrict__ gain,
    const float* __restrict__ w, float* __restrict__ out) {
  __shared__ float lds[256 * 36];
  int tid = blockIdx.x * 256 + threadIdx.x;
  int row = tid >> 10, chunk = tid & (NCHROW - 1);
  const float* gx = x + (size_t)row * TLEN + (size_t)chunk * CHUNKS;
  float* gy = out + (size_t)row * TLEN + (size_t)chunk * CHUNKS;
  uint32_t lds_byte = (uint32_t)(uintptr_t)&lds[threadIdx.x * 36];

  float x1 = 0.f, x2 = 0.f;
  if (chunk > 0) { x1 = gx[-1]; x2 = gx[-2]; }
  async_tile16(lds_byte, gx);

  const float* SY1 = w + 6 * NC + 6 * NPAIR;
  const float* SY2 = SY1 + NC;
  float y1 = SY1[tid], y2 = SY2[tid];
  Ctl ctl = load_ctl(freq, bw, gain, row, chunk);

  for (int tile = 0; tile < 16; ++tile) {
    int buf = tile & 1;
    if (tile < 15) { async_tile16(lds_byte + (buf ^ 1) * 64u, gx + (tile + 1) * 16); wait_async_le4(); }
    else           { wait_async_0(); }
#pragma unroll
    for (int i = 0; i < 16; ++i) {
      float xt = lds[threadIdx.x * 36 + buf * 16 + i];
      float frac = (float)(tile * 16 + i) * (1.0f / 256.0f);
      float b0, a1, a2;
      biquad_coef(ctl, frac, b0, a1, a2);
      float y = b0 * (xt - x2) - a1 * y1 - a2 * y2;
      gy[tile * 16 + i] = y;
      y2 = y1; y1 = y;
      x2 = x1; x1 = xt;
    }
  }
  if (chunk == NCHROW - 1) {                 // final carry (y1, y2, x1, x2)
    size_t bt = (size_t)BATCH * TLEN;
    out[bt + 0 * BATCH + row] = y1;
    out[bt + 1 * BATCH + row] = y2;
    out[bt + 2 * BATCH + row] = x1;
    out[bt + 3 * BATCH + row] = x2;
  }
}

// ---------------------------------------------------------------------------
extern "C" void kernel_launch(void* const* d_in, const int* in_sizes, int n_in,
                              void* d_out, int out_size, void* d_ws, size_t ws_size,
                              hipStream_t stream) {
  (void)in_sizes; (void)n_in; (void)out_size; (void)ws_size;  // ws use: 11*NC floats ~1.44MB
  const float* x    = (const float*)d_in[0];
  const float* freq = (const float*)d_in[1];
  const float* bw   = (const float*)d_in[2];
  const float* gain = (const float*)d_in[3];
  float* out = (float*)d_out;
  float* w   = (float*)d_ws;

  k1_chunk_transforms<<<NC / 256, 256, 0, stream>>>(x, freq, bw, gain, w);
  k2a_pair_compose<<<NPAIR / 16, 256, 0, stream>>>(w);
  k2b_scan<<<BATCH, 32, 0, stream>>>(w);
  k3_finalize<<<NC / 256, 256, 0, stream>>>(x, freq, bw, gain, w, out);
}